// Sampler_45457933861111
// MI455X (gfx1250) — compile-verified
//
#include <hip/hip_runtime.h>
#include <hip/hip_bf16.h>
#include <stdint.h>

// ---------------------------------------------------------------------------
// Top-p sampler for logits (1, 256, 128000) fp32 on MI455X (gfx1250).
//   Pass 1: row max      -- HBM-bound, async global->LDS double buffering
//   Pass 2: sum(exp) + 256-bin log-prob histogram (LDS float atomics),
//           Sigma-exp on the matrix pipe via V_WMMA_F32_16X16X4_F32 with a
//           tied "+v" accumulator (in-place D==C accumulate); unroll disabled
//           so the accumulator tuples stay fixed across the back-edge
//   scan:   histogram prefix scan -> top-p cutoff in (l - max)/T space
//   Pass 3: Gumbel-argmax over the kept set (L2-resident read, prefetch)
//   Finalize: threshold/argmax unmask + scatter into x[start:end]
// ---------------------------------------------------------------------------

typedef float    v2f  __attribute__((ext_vector_type(2)));
typedef float    v8f  __attribute__((ext_vector_type(8)));

#define V_VOCAB 128000
#define VEC4    (V_VOCAB / 4)          // 32000 float4 per row
#define THREADS 1024                   // 32 wave32 per block
#define NFULL   (VEC4 / THREADS)       // 31 uniform chunks
#define TAIL4   (VEC4 - NFULL*THREADS) // 256 float4 tail
#define NROWS   256
#define T_SEQ   2048

#if defined(__HIP_DEVICE_COMPILE__)
// In-place accumulate on the matrix pipe: D and C tied to the same even
// VGPR tuple -> no accumulator copies. D->C reuse back-to-back is not a
// listed WMMA hazard (only D->A/B RAW is), and acc0/acc1 alternate anyway.
#define WMMA_ACC_F32(acc, a, ones)                                         \
  asm("v_wmma_f32_16x16x4_f32 %0, %1, %2, %0"                              \
      : "+v"(acc) : "v"(a), "v"(ones))
#endif

__device__ __forceinline__ float wave_max_f(float v) {
  #pragma unroll
  for (int off = 16; off > 0; off >>= 1)
    v = fmaxf(v, __shfl_xor(v, off, 32));
  return v;
}

__device__ __forceinline__ float wave_sum_f(float v) {
  #pragma unroll
  for (int off = 16; off > 0; off >>= 1)
    v += __shfl_xor(v, off, 32);
  return v;
}

__device__ __forceinline__ int bin_of(float d) {
  // d = (logit - rowmax)/T <= 0.  Bins of width 0.25 in log-prob space,
  // covering d in (-64, 0]; exp(-64) ~ 1.6e-28 is negligible mass.
  int b = (int)(d * -4.0f);
  b = b < 0 ? 0 : b;
  return b > 255 ? 255 : b;
}

__device__ __forceinline__ float gumbel_noise(unsigned row, unsigned idx) {
  unsigned h = idx * 0x9E3779B1u ^ (row * 0x85EBCA77u + 0xC2B2AE3Du);
  h ^= h >> 16; h *= 0x7FEB352Du;
  h ^= h >> 15; h *= 0x846CA68Bu;
  h ^= h >> 16;
  float u = ((float)(h >> 8) + 0.5f) * (1.0f / 16777216.0f);  // (0,1)
  return -__logf(-__logf(u));
}

__global__ __launch_bounds__(THREADS)
void row_sample_kernel(const float* __restrict__ logits,
                       const float* __restrict__ top_p_p,
                       const float* __restrict__ temp_p,
                       int*   __restrict__ ws_idx,
                       float* __restrict__ ws_p) {
  __shared__ __align__(16) float stage[2][THREADS * 4];  // 2 x 16 KB staging
  __shared__ float hist[256];
  __shared__ float redf[32];
  __shared__ int   redi[32];
  __shared__ float bc[4];

  const int tid  = threadIdx.x;
  const int lane = tid & 31;
  const int wv   = tid >> 5;
  const int row  = blockIdx.x;

  const float invT = 1.0f / temp_p[0];
  const float topp = top_p_p[0];
  const float4* rowp = (const float4*)(logits + (size_t)row * V_VOCAB);

  if (tid < 256) hist[tid] = 0.0f;

  // ------------------- Pass 1: row max (async HBM -> LDS) ------------------
  const unsigned lds0 = (unsigned)(uintptr_t)(&stage[0][0]) + (unsigned)tid * 16u;
  const unsigned lds1 = (unsigned)(uintptr_t)(&stage[1][0]) + (unsigned)tid * 16u;

#if defined(__HIP_DEVICE_COMPILE__)
  {
    unsigned long long ga0 = (unsigned long long)(uintptr_t)(rowp + tid);
    asm volatile("global_load_async_to_lds_b128 %0, %1, off"
                 :: "v"(lds0), "v"(ga0) : "memory");
  }
#endif
  float m = -__builtin_inff();
  for (int k = 0; k < NFULL; ++k) {
#if defined(__HIP_DEVICE_COMPILE__)
    if (k + 1 < NFULL) {
      unsigned long long ga =
          (unsigned long long)(uintptr_t)(rowp + (k + 1) * THREADS + tid);
      unsigned dst = ((k + 1) & 1) ? lds1 : lds0;
      asm volatile("global_load_async_to_lds_b128 %0, %1, off"
                   :: "v"(dst), "v"(ga) : "memory");
      asm volatile("s_wait_asynccnt 0x1" ::: "memory");  // chunk k landed
    } else {
      asm volatile("s_wait_asynccnt 0x0" ::: "memory");
    }
#endif
    const float4* sp =
        (const float4*)((k & 1) ? &stage[1][0] : &stage[0][0]) + tid;
    float4 v = *sp;  // each lane reads exactly the 16B it staged: no barrier
    m = fmaxf(m, fmaxf(fmaxf(v.x, v.y), fmaxf(v.z, v.w)));
  }
  if (tid < TAIL4) {
    float4 v = rowp[NFULL * THREADS + tid];
    m = fmaxf(m, fmaxf(fmaxf(v.x, v.y), fmaxf(v.z, v.w)));
  }
  m = wave_max_f(m);
  if (lane == 0) redf[wv] = m;
  __syncthreads();
  if (wv == 0) {
    float t2 = wave_max_f(redf[lane]);
    if (lane == 0) bc[0] = t2;
  }
  __syncthreads();
  m = bc[0];

  // --------- Pass 2: Sigma exp (WMMA) + histogram (L2-resident read) -------
  v8f acc0 = {};            // in-place WMMA accumulators (device path)
  v8f acc1 = {};
  const v2f ones2 = {1.0f, 1.0f};
  float fsum = 0.0f;        // host-parse accumulator
  (void)fsum;

  // No unrolling: keeps each accumulator pinned to one even VGPR tuple
  // across the back-edge (unroll x2 forced tuple rotation + mov chains).
  #pragma clang loop unroll(disable)
  for (int k = 0; k < NFULL; ++k) {   // trip count uniform -> EXEC all ones
    __builtin_prefetch(rowp + (k + 2) * THREADS + tid, 0, 3);
    float4 v = rowp[k * THREADS + tid];
    float d0 = (v.x - m) * invT, d1 = (v.y - m) * invT;
    float d2 = (v.z - m) * invT, d3 = (v.w - m) * invT;
    float e0 = __expf(d0), e1 = __expf(d1), e2 = __expf(d2), e3 = __expf(d3);
    atomicAdd(&hist[bin_of(d0)], e0);
    atomicAdd(&hist[bin_of(d1)], e1);
    atomicAdd(&hist[bin_of(d2)], e2);
    atomicAdd(&hist[bin_of(d3)], e3);
#if defined(__HIP_DEVICE_COMPILE__)
    // D = A x ones + C: each column accumulates the wave-wide sum of the
    // 64 f32 fed per instruction (matrix-pipe reduction, D==C in place).
    v2f a0 = {e0, e1};
    v2f a1 = {e2, e3};
    WMMA_ACC_F32(acc0, a0, ones2);
    WMMA_ACC_F32(acc1, a1, ones2);
#else
    fsum += (e0 + e1) + (e2 + e3);
#endif
  }
  float tailsum = 0.0f;
  if (tid < TAIL4) {   // divergent tail: scalar path only (no WMMA here)
    float4 v = rowp[NFULL * THREADS + tid];
    float d0 = (v.x - m) * invT, d1 = (v.y - m) * invT;
    float d2 = (v.z - m) * invT, d3 = (v.w - m) * invT;
    float e0 = __expf(d0), e1 = __expf(d1), e2 = __expf(d2), e3 = __expf(d3);
    atomicAdd(&hist[bin_of(d0)], e0);
    atomicAdd(&hist[bin_of(d1)], e1);
    atomicAdd(&hist[bin_of(d2)], e2);
    atomicAdd(&hist[bin_of(d3)], e3);
    tailsum = (e0 + e1) + (e2 + e3);
  }

#if defined(__HIP_DEVICE_COMPILE__)
  float sigma = (acc0[0] + acc0[1] + acc0[2] + acc0[3] +
                 acc0[4] + acc0[5] + acc0[6] + acc0[7]) +
                (acc1[0] + acc1[1] + acc1[2] + acc1[3] +
                 acc1[4] + acc1[5] + acc1[6] + acc1[7]);
#else
  float sigma = fsum;   // host compile pass only; never executed
#endif
  // sigma holds one column's partial (rows 0-7 or 8-15); pair up across the
  // half-wave boundary to get the full wave sum in every lane.
  float colsum = sigma + __shfl_xor(sigma, 16, 32);
  float wtotal = colsum + wave_sum_f(tailsum);
  if (lane == 0) redf[wv] = wtotal;
  __syncthreads();

  if (wv == 0) {
    float z = wave_sum_f(redf[lane]);
    if (lane == 0) {
      float targ = topp * z;
      float csum = 0.0f;
      int kbin = 255;
      for (int b2 = 0; b2 < 256; ++b2) {
        csum += hist[b2];
        if (csum >= targ) { kbin = b2; break; }
      }
      bc[1] = -(float)(kbin + 1) * 0.25f;  // cutoff in d-space
      bc[2] = csum;                        // Z over kept set
    }
  }
  __syncthreads();
  const float cutoff = bc[1];
  const float zkept  = bc[2];

  // ----------------- Pass 3: Gumbel-argmax over kept set -------------------
  float best = -__builtin_inff();
  int   bidx = 0;
  for (int i = tid; i < VEC4; i += THREADS) {
    __builtin_prefetch(rowp + i + THREADS, 0, 3);
    float4 v = rowp[i];
    int base = i * 4;
    float d0 = (v.x - m) * invT, d1 = (v.y - m) * invT;
    float d2 = (v.z - m) * invT, d3 = (v.w - m) * invT;
    if (d0 >= cutoff) { float t2 = d0 + gumbel_noise(row, base + 0); if (t2 > best) { best = t2; bidx = base + 0; } }
    if (d1 >= cutoff) { float t2 = d1 + gumbel_noise(row, base + 1); if (t2 > best) { best = t2; bidx = base + 1; } }
    if (d2 >= cutoff) { float t2 = d2 + gumbel_noise(row, base + 2); if (t2 > best) { best = t2; bidx = base + 2; } }
    if (d3 >= cutoff) { float t2 = d3 + gumbel_noise(row, base + 3); if (t2 > best) { best = t2; bidx = base + 3; } }
  }
  #pragma unroll
  for (int off = 16; off > 0; off >>= 1) {
    float ov = __shfl_xor(best, off, 32);
    int   oi = __shfl_xor(bidx, off, 32);
    if (ov > best) { best = ov; bidx = oi; }
  }
  if (lane == 0) { redf[wv] = best; redi[wv] = bidx; }
  __syncthreads();
  if (tid == 0) {
    float bb = redf[0];
    int   bi = redi[0];
    for (int w2 = 1; w2 < 32; ++w2)
      if (redf[w2] > bb) { bb = redf[w2]; bi = redi[w2]; }
    float dsel = (logits[(size_t)row * V_VOCAB + bi] - m) * invT;
    ws_idx[row] = bi;
    ws_p[row]   = __expf(dsel) / zkept;
  }
}

__global__ __launch_bounds__(256)
void finalize_kernel(const int* __restrict__ x,
                     const int* __restrict__ ws_idx,
                     const float* __restrict__ ws_p,
                     const float* __restrict__ thr_p,
                     const unsigned char* __restrict__ mask,
                     const int* __restrict__ start_p,
                     const int* __restrict__ end_p,
                     int* __restrict__ out) {
  __shared__ float rv[256];
  __shared__ int   ri[256];
  __shared__ int   xblk[256];

  const int t     = threadIdx.x;
  const int start = start_p[0];
  const int end   = end_p[0];
  const int blk   = end - start;          // 256 for this problem
  const float thr = thr_p[0];

  float p = -__builtin_inff();
  unsigned char mk = 0;
  if (t < blk) {
    mk = mask[start + t];
    p  = mk ? ws_p[t] : -__builtin_inff();
  }
  rv[t] = p; ri[t] = t;
  __syncthreads();
  for (int off = 128; off > 0; off >>= 1) {
    if (t < off && rv[t + off] > rv[t]) { rv[t] = rv[t + off]; ri[t] = ri[t + off]; }
    __syncthreads();
  }
  const int max_row = ri[0];
  const bool un = ((p > thr) || (t == max_row)) && (mk != 0);
  if (t < blk) xblk[t] = un ? ws_idx[t] : x[start + t];
  __syncthreads();
  for (int pos = t; pos < T_SEQ; pos += 256) {
    out[pos] = (pos >= start && pos < end) ? xblk[pos - start] : x[pos];
  }
}

extern "C" void kernel_launch(void* const* d_in, const int* in_sizes, int n_in,
                              void* d_out, int out_size, void* d_ws, size_t ws_size,
                              hipStream_t stream) {
  (void)in_sizes; (void)n_in; (void)out_size; (void)ws_size;
  const int*   x      = (const int*)d_in[0];
  const float* logits = (const float*)d_in[1];
  const float* top_p  = (const float*)d_in[2];
  const float* temp   = (const float*)d_in[3];
  const float* thr    = (const float*)d_in[4];
  const unsigned char* mask = (const unsigned char*)d_in[5];
  const int*   startp = (const int*)d_in[6];
  const int*   endp   = (const int*)d_in[7];

  int*   ws_idx = (int*)d_ws;                       // 256 ints
  float* ws_p   = (float*)((char*)d_ws + 1024);     // 256 floats

  row_sample_kernel<<<NROWS, THREADS, 0, stream>>>(logits, top_p, temp,
                                                   ws_idx, ws_p);
  finalize_kernel<<<1, 256, 0, stream>>>(x, ws_idx, ws_p, thr, mask,
                                         startp, endp, (int*)d_out);
}